// TriLinearSimilarity_35407710388742
// MI455X (gfx1250) — compile-verified
//
#include <hip/hip_runtime.h>

// CDNA5 / gfx1250, wave32. Memory-bound streaming GEMV folded into
// V_WMMA_F32_16X16X4_F32: out[s] = t1[s,:]·w1 + t2[s,:]·w2 + (t1∘t2)[s,:]·w3 + bias.
// A = weight chunk replicated over M rows; B = 16 output rows' data.
// K is processed in super-chunks of 8 with a lane-half permutation so every
// operand fragment pair comes from one global_load_b128 (16 B/lane, 32 B
// contiguous per row per request).

typedef __attribute__((ext_vector_type(2))) float v2f;
typedef __attribute__((ext_vector_type(4))) float v4f;
typedef __attribute__((ext_vector_type(8))) float v8f;

#define TL_D 128
#define WAVES_PER_BLOCK 8
#define OUTS_PER_WAVE 16

static __device__ __forceinline__ v2f lo2(v4f v) { return __builtin_shufflevector(v, v, 0, 1); }
static __device__ __forceinline__ v2f hi2(v4f v) { return __builtin_shufflevector(v, v, 2, 3); }

__global__ __launch_bounds__(256) void TriLinearSimilarity_wmma_kernel(
    const float* __restrict__ t1,
    const float* __restrict__ t2,
    const float* __restrict__ w,
    const float* __restrict__ bias,
    float* __restrict__ out,
    int total_outputs)
{
    const int lane   = threadIdx.x & 31;
    const int waveId = threadIdx.x >> 5;

    // 16 consecutive flat output positions per wave.
    const long tile  = (long)blockIdx.x * WAVES_PER_BLOCK + waveId;
    const long sBase = tile * OUTS_PER_WAVE;
    if (sBase >= (long)total_outputs) return;   // wave-uniform, EXEC stays all-1s

    const int laneN  = lane & 15;        // output column (N) this lane owns
    const int laneHi = lane >> 4;        // which K-half of the fragment this lane feeds
    const int kOff   = 4 * laneHi;       // byte-level: +16B for upper half lanes

    // Row base pointers for this lane's output position.
    const float* __restrict__ r1 = t1 + (sBase + laneN) * (long)TL_D;
    const float* __restrict__ r2 = t2 + (sBase + laneN) * (long)TL_D;

    v8f acc = {};

    #pragma unroll 2
    for (int k0 = 0; k0 < TL_D; k0 += 8) {
        const int off = k0 + kOff;

        // A fragments (weights, replicated over M -> broadcast loads, L0 hits).
        v4f W1 = *(const v4f*)(w            + off);
        v4f W2 = *(const v4f*)(w + TL_D     + off);
        v4f W3 = *(const v4f*)(w + 2 * TL_D + off);

        // B fragments (streamed data): one b128 per tensor per super-chunk.
        v4f B1 = *(const v4f*)(r1 + off);
        v4f B2 = *(const v4f*)(r2 + off);
        v4f BP = B1 * B2;   // elementwise product term (2x v_pk_mul_f32)

        // Super-chunk slot 0: K values {k0,k0+1 | k0+4,k0+5}
        acc = __builtin_amdgcn_wmma_f32_16x16x4_f32(
                  false, lo2(W1), false, lo2(B1), (short)0, acc, false, false);
        acc = __builtin_amdgcn_wmma_f32_16x16x4_f32(
                  false, lo2(W2), false, lo2(B2), (short)0, acc, false, false);
        acc = __builtin_amdgcn_wmma_f32_16x16x4_f32(
                  false, lo2(W3), false, lo2(BP), (short)0, acc, false, false);
        // Super-chunk slot 1: K values {k0+2,k0+3 | k0+6,k0+7}
        acc = __builtin_amdgcn_wmma_f32_16x16x4_f32(
                  false, hi2(W1), false, hi2(B1), (short)0, acc, false, false);
        acc = __builtin_amdgcn_wmma_f32_16x16x4_f32(
                  false, hi2(W2), false, hi2(B2), (short)0, acc, false, false);
        acc = __builtin_amdgcn_wmma_f32_16x16x4_f32(
                  false, hi2(W3), false, hi2(BP), (short)0, acc, false, false);
    }

    // All M rows of D are identical; row M=0 lives in acc[0], lanes 0-15 (N = lane).
    if (lane < 16) {
        const float r = acc[0] + bias[0];
        __builtin_nontemporal_store(r, &out[sBase + lane]);
    }
}

extern "C" void kernel_launch(void* const* d_in, const int* in_sizes, int n_in,
                              void* d_out, int out_size, void* d_ws, size_t ws_size,
                              hipStream_t stream) {
    const float* t1   = (const float*)d_in[0];   // (4,512,512,128) f32
    const float* t2   = (const float*)d_in[1];   // (4,512,512,128) f32
    const float* w    = (const float*)d_in[2];   // (384,) f32
    const float* bias = (const float*)d_in[3];   // (1,)  f32
    float* out        = (float*)d_out;           // (4,512,512) f32

    const int total  = out_size;                               // 1,048,576
    const int tiles  = (total + OUTS_PER_WAVE - 1) / OUTS_PER_WAVE;
    const int blocks = (tiles + WAVES_PER_BLOCK - 1) / WAVES_PER_BLOCK;

    TriLinearSimilarity_wmma_kernel<<<blocks, 256, 0, stream>>>(
        t1, t2, w, bias, out, total);
}